// NSGAT_30691836297930
// MI455X (gfx1250) — compile-verified
//
#include <hip/hip_runtime.h>

typedef float v2f __attribute__((ext_vector_type(2)));
typedef float v8f __attribute__((ext_vector_type(8)));

#define NEG_SLOPE 0.2f

#if defined(__has_builtin)
#if __has_builtin(__builtin_amdgcn_global_load_async_to_lds_b32) && \
    __has_builtin(__builtin_amdgcn_s_wait_asynccnt)
#define USE_ASYNC_LDS 1
#endif
#endif
#ifndef USE_ASYNC_LDS
#define USE_ASYNC_LDS 0
#endif

typedef int __attribute__((address_space(1))) as1_int;
typedef int __attribute__((address_space(3))) as3_int;

// Monotone order-preserving float<->uint mapping so we can do float-max with
// integer atomicMax (works for negative values).
__device__ __forceinline__ unsigned fenc(float x) {
  unsigned u = __float_as_uint(x);
  return (u & 0x80000000u) ? ~u : (u | 0x80000000u);
}
__device__ __forceinline__ float fdec(unsigned k) {
  return __uint_as_float((k & 0x80000000u) ? (k ^ 0x80000000u) : ~k);
}
#define FENC_NEG_INF 0x007FFFFFu  // fenc(-inf)

// ---------------------------------------------------------------------------
// C(NxM) = A(NxK) * B(KxM), fp32, V_WMMA_F32_16X16X4_F32.
// Block = 256 threads = 8 waves; block tile = 128 rows x 64 cols.
// K is processed in chunks of KC=128: the B chunk (KC x 64) is staged once per
// block into LDS (async global->LDS when available), transposed with a
// +4-float pad so each lane's (k,k+1) pair is one bank-conflict-free
// ds_load_b64.  A is a pointer-incremented global_load_b64 per lane.  All
// index guards are clamps (valid addresses, results discarded at the guarded
// store) so EXEC is all-ones at every WMMA.
// A frag (16x4 f32): lane l -> row (l&15), K-pair ((l>>4)*2, +1).
// B frag (4x16 f32): lane l -> col (l&15), K-pair ((l>>4)*2, +1).
// D (16x16 f32):     vgpr i -> row i + (l<16 ? 0 : 8), col l&15.
// ---------------------------------------------------------------------------
#define GEMM_KC 128
#define GEMM_LB (GEMM_KC + 4)  // padded LDS stride (floats)

__global__ __launch_bounds__(256) void gemm_wmma_f32(
    const float* __restrict__ A, const float* __restrict__ B,
    float* __restrict__ C, int N, int K, int M) {
  __shared__ float sB[64 * GEMM_LB];  // 33,792 B

  const int tid = threadIdx.x;
  const int lane = tid & 31;
  const int wave = tid >> 5;
  const int rowBase = blockIdx.x * 128 + wave * 16;
  const int colBase = blockIdx.y * 64;
  const int mrow = lane & 15;
  const int khalf = (lane >> 4) << 1;  // 0 or 2

  int arow = rowBase + mrow;
  if (arow >= N) arow = N - 1;
  const float* pA = A + (size_t)arow * K + khalf;

  // staging coordinates: thread t covers column (t&63), rows (t>>6) + 4*i
  const int sc = tid & 63;
  const int sk0 = tid >> 6;
  int sgc = colBase + sc;
  if (sgc >= M) sgc = M - 1;

  v8f acc0 = {}, acc1 = {}, acc2 = {}, acc3 = {};

  for (int kc = 0; kc < K; kc += GEMM_KC) {
    // ---- stage B[kc..kc+KC) x [colBase..colBase+64) transposed into LDS ----
    {
      const float* gB = B + (size_t)(kc + sk0) * M + sgc;
      int lidx = sc * GEMM_LB + sk0;
      #pragma unroll 4
      for (int kk = sk0; kk < GEMM_KC; kk += 4) {
#if USE_ASYNC_LDS
        __builtin_amdgcn_global_load_async_to_lds_b32(
            (as1_int*)(float*)gB, (as3_int*)&sB[lidx], 0, 0);
#else
        sB[lidx] = *gB;
#endif
        gB += (size_t)4 * M;
        lidx += 4;
      }
#if USE_ASYNC_LDS
      __builtin_amdgcn_s_wait_asynccnt(0);
#endif
    }
    __syncthreads();

    // ---- compute over the staged chunk ----
    #pragma unroll 4
    for (int k = 0; k < GEMM_KC; k += 4) {
      v2f a = *(const v2f*)pA;
      pA += 4;
      const int kk = k + khalf;
      v2f b0 = *(const v2f*)&sB[(mrow + 0) * GEMM_LB + kk];
      v2f b1 = *(const v2f*)&sB[(mrow + 16) * GEMM_LB + kk];
      v2f b2 = *(const v2f*)&sB[(mrow + 32) * GEMM_LB + kk];
      v2f b3 = *(const v2f*)&sB[(mrow + 48) * GEMM_LB + kk];
      acc0 = __builtin_amdgcn_wmma_f32_16x16x4_f32(false, a, false, b0, (short)0, acc0, false, false);
      acc1 = __builtin_amdgcn_wmma_f32_16x16x4_f32(false, a, false, b1, (short)0, acc1, false, false);
      acc2 = __builtin_amdgcn_wmma_f32_16x16x4_f32(false, a, false, b2, (short)0, acc2, false, false);
      acc3 = __builtin_amdgcn_wmma_f32_16x16x4_f32(false, a, false, b3, (short)0, acc3, false, false);
    }
    __syncthreads();
  }

  const int rOff = (lane >> 4) << 3;  // lanes 16-31 hold rows +8..+15
  #pragma unroll
  for (int i = 0; i < 8; ++i) {
    int r = rowBase + rOff + i;
    if (r < N) {
      float* Crow = C + (size_t)r * M;
      int g0 = colBase + 0 + mrow;  if (g0 < M) Crow[g0] = acc0[i];
      int g1 = colBase + 16 + mrow; if (g1 < M) Crow[g1] = acc1[i];
      int g2 = colBase + 32 + mrow; if (g2 < M) Crow[g2] = acc2[i];
      int g3 = colBase + 48 + mrow; if (g3 < M) Crow[g3] = acc3[i];
    }
  }
}

// el[n,h] = f[n,h,:]·al[h,:],  er[n,h] = f[n,h,:]·ar[h,:]
template <int H, int D>
__global__ __launch_bounds__(256) void attn_scores_k(
    const float* __restrict__ f, const float* __restrict__ al,
    const float* __restrict__ ar, float* __restrict__ el,
    float* __restrict__ er, int N) {
  int i = blockIdx.x * blockDim.x + threadIdx.x;
  if (i >= N * H) return;
  int n = i / H, h = i % H;
  const float* fr = f + (size_t)n * (H * D) + h * D;
  float sl = 0.f, sr = 0.f;
  #pragma unroll 4
  for (int d = 0; d < D; ++d) {
    float v = fr[d];
    sl += v * al[h * D + d];
    sr += v * ar[h * D + d];
  }
  el[i] = sl;
  er[i] = sr;
}

__global__ __launch_bounds__(256) void init_ms(unsigned* __restrict__ mk,
                                               float* __restrict__ s, int NH) {
  int i = blockIdx.x * blockDim.x + threadIdx.x;
  if (i < NH) { mk[i] = FENC_NEG_INF; s[i] = 0.0f; }
}

// Seed out with the bias so the edge scatter-add lands on (segment_sum + b).
__global__ __launch_bounds__(256) void init_out_bias(float* __restrict__ out,
                                                     const float* __restrict__ b,
                                                     long long total, int HD) {
  long long i = (long long)blockIdx.x * blockDim.x + threadIdx.x;
  if (i < total) out[i] = b[(int)(i % HD)];
}

template <int H>
__global__ __launch_bounds__(256) void edge_max_k(
    const int* __restrict__ src, const int* __restrict__ dst,
    const float* __restrict__ el, const float* __restrict__ er,
    unsigned* __restrict__ mk, int E) {
  int i = blockIdx.x * blockDim.x + threadIdx.x;
  if (i >= E * H) return;
  int e = i / H, h = i % H;
  int dn = dst[e];
  float v = el[(size_t)src[e] * H + h] + er[(size_t)dn * H + h];
  v = v > 0.f ? v : NEG_SLOPE * v;
  atomicMax(&mk[(size_t)dn * H + h], fenc(v));
}

template <int H>
__global__ __launch_bounds__(256) void edge_exp_k(
    const int* __restrict__ src, const int* __restrict__ dst,
    const float* __restrict__ el, const float* __restrict__ er,
    const unsigned* __restrict__ mk, float* __restrict__ ex,
    float* __restrict__ s, int E) {
  int i = blockIdx.x * blockDim.x + threadIdx.x;
  if (i >= E * H) return;
  int e = i / H, h = i % H;
  int dn = dst[e];
  float v = el[(size_t)src[e] * H + h] + er[(size_t)dn * H + h];
  v = v > 0.f ? v : NEG_SLOPE * v;
  float m = fdec(mk[(size_t)dn * H + h]);
  float x = __expf(v - m);
  ex[i] = x;
  atomicAdd(&s[(size_t)dn * H + h], x);
}

// out[dst,h,d] += (ex/s)[e,h] * f[src,h,d]; consecutive lanes cover
// consecutive d -> coalesced gather & scatter; f (51 MB) is L2-resident.
template <int H, int D>
__global__ __launch_bounds__(256) void edge_agg_k(
    const int* __restrict__ src, const int* __restrict__ dst,
    const float* __restrict__ ex, const float* __restrict__ s,
    const float* __restrict__ f, float* __restrict__ out, int E) {
  long long idx = (long long)blockIdx.x * blockDim.x + threadIdx.x;
  long long total = (long long)E * H * D;
  if (idx >= total) return;
  int d = (int)(idx % D);
  long long t = idx / D;
  int h = (int)(t % H);
  int e = (int)(t / H);
  int sn = src[e], dn = dst[e];
  if (e + 16 < E) {  // warm L2/WGP$ for an upcoming edge's source row
    int sp = src[e + 16];
    __builtin_prefetch(&f[(size_t)sp * (H * D) + h * D + d], 0, 1);
  }
  float a = ex[(size_t)e * H + h] / s[(size_t)dn * H + h];
  atomicAdd(&out[(size_t)dn * (H * D) + h * D + d],
            a * f[(size_t)sn * (H * D) + h * D + d]);
}

template <int H, int D>
static void run_layer(const float* h, int K, const float* W, const float* al,
                      const float* ar, const float* bias, int N, int E,
                      const int* src, const int* dst, float* f, float* out,
                      float* el, float* er, unsigned* mk, float* s, float* ex,
                      hipStream_t stream) {
  const int M = H * D;
  dim3 gg((N + 127) / 128, (M + 63) / 64);
  gemm_wmma_f32<<<gg, 256, 0, stream>>>(h, W, f, N, K, M);

  int nh = N * H;
  attn_scores_k<H, D><<<(nh + 255) / 256, 256, 0, stream>>>(f, al, ar, el, er, N);
  init_ms<<<(nh + 255) / 256, 256, 0, stream>>>(mk, s, nh);

  long long ntot = (long long)N * M;
  init_out_bias<<<(unsigned)((ntot + 255) / 256), 256, 0, stream>>>(out, bias, ntot, M);

  int eh = E * H;
  edge_max_k<H><<<(eh + 255) / 256, 256, 0, stream>>>(src, dst, el, er, mk, E);
  edge_exp_k<H><<<(eh + 255) / 256, 256, 0, stream>>>(src, dst, el, er, mk, ex, s, E);

  long long etot = (long long)E * H * D;
  edge_agg_k<H, D><<<(unsigned)((etot + 255) / 256), 256, 0, stream>>>(src, dst, ex, s, f, out, E);
}

extern "C" void kernel_launch(void* const* d_in, const int* in_sizes, int n_in,
                              void* d_out, int out_size, void* d_ws, size_t ws_size,
                              hipStream_t stream) {
  const float* x   = (const float*)d_in[0];
  const int*   src = (const int*)d_in[1];
  const int*   dst = (const int*)d_in[2];
  const float* W0  = (const float*)d_in[3];
  const float* al0 = (const float*)d_in[4];
  const float* ar0 = (const float*)d_in[5];
  const float* b0  = (const float*)d_in[6];
  const float* W1  = (const float*)d_in[7];
  const float* al1 = (const float*)d_in[8];
  const float* ar1 = (const float*)d_in[9];
  const float* b1  = (const float*)d_in[10];
  const float* W2  = (const float*)d_in[11];
  const float* al2 = (const float*)d_in[12];
  const float* ar2 = (const float*)d_in[13];
  const float* b2  = (const float*)d_in[14];

  const int IN = 128, H = 4, D = 64, HD = 256;
  const int N = in_sizes[0] / IN;
  const int E = in_sizes[1];

  char* w = (char*)d_ws;
  size_t off = 0;
  auto carve = [&](size_t bytes) -> void* {
    void* p = w + off;
    off += (bytes + 255) & ~(size_t)255;
    return p;
  };
  float*    bufF = (float*)carve((size_t)N * HD * sizeof(float));  // f of current layer
  float*    bufA = (float*)carve((size_t)N * HD * sizeof(float));  // layer0 out / layer1 in
  float*    bufB = (float*)carve((size_t)N * HD * sizeof(float));  // layer1 out / layer2 in
  float*    el   = (float*)carve((size_t)N * H * sizeof(float));
  float*    er   = (float*)carve((size_t)N * H * sizeof(float));
  unsigned* mk   = (unsigned*)carve((size_t)N * H * sizeof(unsigned));
  float*    s    = (float*)carve((size_t)N * H * sizeof(float));
  float*    ex   = (float*)carve((size_t)E * H * sizeof(float));

  run_layer<4, 64>(x,    IN, W0, al0, ar0, b0, N, E, src, dst, bufF, bufA, el, er, mk, s, ex, stream);
  run_layer<4, 64>(bufA, HD, W1, al1, ar1, b1, N, E, src, dst, bufF, bufB, el, er, mk, s, ex, stream);
  // Layer 2: H=1, D=C=47; mean over a single head is identity, write d_out.
  run_layer<1, 47>(bufB, HD, W2, al2, ar2, b2, N, E, src, dst, bufF, (float*)d_out, el, er, mk, s, ex, stream);

  (void)n_in; (void)out_size; (void)ws_size;
}